// NodeNet_21397527069365
// MI455X (gfx1250) — compile-verified
//
#include <hip/hip_runtime.h>
#include <hip/hip_bf16.h>

typedef float v2f __attribute__((ext_vector_type(2)));
typedef float v8f __attribute__((ext_vector_type(8)));

#define NQ 12
#define NN 2048
#define EE 4096
#define N_OPS 34

// Circuit: kind 0 = u3(theta[a],theta[a+1],theta[a+2]) on wire b; kind 1 = CNOT(control=a, target=b)
__constant__ int c_kind[N_OPS] = {0,0,1, 0,0,1, 0,0,1, 0,0,1, 0,0,1, 0,0,1,
                                  0,0,1, 0,0,1, 0,0,1, 0,0,1, 0,0,1, 0};
__constant__ int c_a[N_OPS]    = {0,3,0, 6,9,3, 12,15,4, 18,21,7, 24,27,8, 30,33,11,
                                  36,39,1, 42,45,6, 48,51,10, 54,57,2, 60,63,5, 66};
__constant__ int c_b[N_OPS]    = {0,1,1, 2,3,2, 4,5,5, 6,7,6, 8,9,9, 10,11,10,
                                  1,2,2, 5,6,5, 9,10,9, 2,5,5, 5,9,9, 4};

// ---------------------------------------------------------------------------
// Kernel 1: bo = Ro^T X, bi = Ri^T X  via V_WMMA_F32_16X16X4_F32
// One wave per 16-row e-tile. A[m=e_local,k=n_local] = R[n,e]; B[k,n_col]=Xpad.
// Zero-padding of B columns 4..15 done by arithmetic mask (no exec divergence).
// blockIdx.y: 0 -> (Ro, bo), 1 -> (Ri, bi)
// ---------------------------------------------------------------------------
__global__ __launch_bounds__(32) void gemm_bt_kernel(
    const float* __restrict__ Ro, const float* __restrict__ Ri,
    const float* __restrict__ X, float* __restrict__ bo, float* __restrict__ bi)
{
    const float* R = (blockIdx.y == 0) ? Ro : Ri;
    float* Bout    = (blockIdx.y == 0) ? bo : bi;
    const int lane = threadIdx.x;
    const int r16  = lane & 15;       // M row (e_local) for A / N col (d) for B
    const int kh   = lane >> 4;       // which K half-pair this lane holds
    const int e0   = blockIdx.x * 16 + r16;
    const float cmask = (r16 < 4) ? 1.0f : 0.0f;  // zero-pad cols 4..15
    const int   cidx  = r16 & 3;                  // clamped X column
    v8f acc = {};
#pragma unroll 4
    for (int n = 0; n < NN; n += 4) {
        const int k0 = n + 2 * kh;
        v2f A, Bv;
        A.x  = R[(size_t)k0 * EE + e0];
        A.y  = R[(size_t)(k0 + 1) * EE + e0];
        Bv.x = X[k0 * 4 + cidx] * cmask;
        Bv.y = X[(k0 + 1) * 4 + cidx] * cmask;
        acc = __builtin_amdgcn_wmma_f32_16x16x4_f32(
            false, A, false, Bv, (short)0, acc, false, false);
    }
    if (r16 < 4) {
        const int ebase = blockIdx.x * 16 + 8 * kh;
#pragma unroll
        for (int v = 0; v < 8; ++v)
            Bout[(ebase + v) * 4 + r16] = acc[v];
    }
}

// ---------------------------------------------------------------------------
// Kernel 2: mi = (Ri*e) bo, mo = (Ro*e) bi -> Mbuf (2048 x 8), cols 0..3 = mi,
// cols 4..7 = mo. One wave per 16-row n-tile. blockIdx.y selects mi/mo.
// A pair and evec pair are contiguous (k0 even) -> float2 (b64) loads.
// ---------------------------------------------------------------------------
__global__ __launch_bounds__(32) void gemm_m_kernel(
    const float* __restrict__ Ri, const float* __restrict__ Ro,
    const float* __restrict__ evec,
    const float* __restrict__ bo, const float* __restrict__ bi,
    float* __restrict__ Mbuf)
{
    const float* R  = (blockIdx.y == 0) ? Ri : Ro;
    const float* Bm = (blockIdx.y == 0) ? bo : bi;
    const int lane = threadIdx.x;
    const int r16  = lane & 15;
    const int kh   = lane >> 4;
    const int nrow = blockIdx.x * 16 + r16;   // A-matrix M row (sample index)
    const float cmask = (r16 < 4) ? 1.0f : 0.0f;
    const int   cidx  = r16 & 3;
    const float* Rrow = R + (size_t)nrow * EE;
    v8f acc = {};
#pragma unroll 4
    for (int e = 0; e < EE; e += 4) {
        const int k0 = e + 2 * kh;
        const float2 av = *(const float2*)(Rrow + k0);
        const float2 ev = *(const float2*)(evec + k0);
        v2f A, Bv;
        A.x  = av.x * ev.x;
        A.y  = av.y * ev.y;
        Bv.x = Bm[k0 * 4 + cidx] * cmask;
        Bv.y = Bm[(k0 + 1) * 4 + cidx] * cmask;
        acc = __builtin_amdgcn_wmma_f32_16x16x4_f32(
            false, A, false, Bv, (short)0, acc, false, false);
    }
    if (r16 < 4) {
        const int nbase = blockIdx.x * 16 + 8 * kh;
        const int col   = blockIdx.y * 4 + r16;
#pragma unroll
        for (int v = 0; v < 8; ++v)
            Mbuf[(nbase + v) * 8 + col] = acc[v];
    }
}

// ---------------------------------------------------------------------------
// Kernel 3: 12-qubit statevector sim, one workgroup (256 thr) per sample.
// State (4096 complex) in LDS (33 KB of the 320 KB WGP pool -> ~9 blocks/WGP).
// Qubit q -> bit (11-q). Wire 9 -> bit 2.
// ---------------------------------------------------------------------------
__global__ __launch_bounds__(256) void sim_kernel(
    const float* __restrict__ Mbuf, const float* __restrict__ X,
    const float* __restrict__ theta, float* __restrict__ out)
{
    __shared__ float sre[4096];
    __shared__ float sim_[4096];
    __shared__ float red[256];

    const int n   = blockIdx.x;
    const int tid = threadIdx.x;

    // angles: M = [mi | mo | X]
    float cq[NQ], sq[NQ];
#pragma unroll
    for (int q = 0; q < NQ; ++q) {
        float ang = (q < 8) ? Mbuf[n * 8 + q] : X[n * 4 + (q - 8)];
        sincosf(0.5f * ang, &sq[q], &cq[q]);
    }

    // product state init: amp[b] = prod_q (bit==0 ? cos : sin), imag = 0
    for (int b = tid; b < 4096; b += 256) {
        float a = 1.0f;
#pragma unroll
        for (int q = 0; q < NQ; ++q)
            a *= ((b >> (11 - q)) & 1) ? sq[q] : cq[q];
        sre[b] = a;
        sim_[b] = 0.0f;
    }
    __syncthreads();

    for (int op = 0; op < N_OPS; ++op) {
        if (c_kind[op] == 0) {
            // ---- u3 gate on wire w ----
            const int a = c_a[op];
            const int p = 11 - c_b[op];
            const int mask = 1 << p;
            const float th = theta[a], ph = theta[a + 1], la = theta[a + 2];
            float st, ct, sl, cl, sp, cp, spl, cpl;
            sincosf(0.5f * th, &st, &ct);
            sincosf(la, &sl, &cl);
            sincosf(ph, &sp, &cp);
            sincosf(ph + la, &spl, &cpl);
            // U00 = ct (real); U01 = -e^{i la} st; U10 = e^{i ph} st; U11 = e^{i(ph+la)} ct
            const float u01r = -cl * st, u01i = -sl * st;
            const float u10r =  cp * st, u10i =  sp * st;
            const float u11r = cpl * ct, u11i = spl * ct;
            for (int j = tid; j < 2048; j += 256) {
                const int i0 = ((j >> p) << (p + 1)) | (j & (mask - 1));
                const int i1 = i0 | mask;
                const float r0 = sre[i0], q0 = sim_[i0];
                const float r1 = sre[i1], q1 = sim_[i1];
                sre[i0]  = ct * r0 + u01r * r1 - u01i * q1;
                sim_[i0] = ct * q0 + u01r * q1 + u01i * r1;
                sre[i1]  = u10r * r0 - u10i * q0 + u11r * r1 - u11i * q1;
                sim_[i1] = u10r * q0 + u10i * r0 + u11r * q1 + u11i * r1;
            }
        } else {
            // ---- CNOT(control=a, target=b): swap amps where c=1, t=0 <-> t=1 ----
            const int pc = 11 - c_a[op];
            const int pt = 11 - c_b[op];
            const int pl = (pc < pt) ? pc : pt;
            const int ph2 = (pc < pt) ? pt : pc;
            for (int j = tid; j < 1024; j += 256) {
                int t1 = ((j >> pl) << (pl + 1)) | (j & ((1 << pl) - 1));
                int t2 = ((t1 >> ph2) << (ph2 + 1)) | (t1 & ((1 << ph2) - 1));
                const int i0 = t2 | (1 << pc);       // c=1, t=0
                const int i1 = i0 | (1 << pt);       // c=1, t=1
                const float tr = sre[i0], ti = sim_[i0];
                sre[i0] = sre[i1];  sim_[i0] = sim_[i1];
                sre[i1] = tr;       sim_[i1] = ti;
            }
        }
        __syncthreads();
    }

    // <Z_9>: sign by bit 2
    float acc = 0.0f;
    for (int b = tid; b < 4096; b += 256) {
        const float pr = sre[b] * sre[b] + sim_[b] * sim_[b];
        acc += ((b >> 2) & 1) ? -pr : pr;
    }
    red[tid] = acc;
    __syncthreads();
    for (int s = 128; s > 0; s >>= 1) {
        if (tid < s) red[tid] += red[tid + s];
        __syncthreads();
    }
    if (tid == 0)
        out[n] = 3.14159265358979323846f * (1.0f - red[0]);
}

// ---------------------------------------------------------------------------
extern "C" void kernel_launch(void* const* d_in, const int* in_sizes, int n_in,
                              void* d_out, int out_size, void* d_ws, size_t ws_size,
                              hipStream_t stream) {
    (void)in_sizes; (void)n_in; (void)out_size; (void)ws_size;
    const float* X     = (const float*)d_in[0];  // (2048, 4)
    const float* evec  = (const float*)d_in[1];  // (4096,)
    const float* Ri    = (const float*)d_in[2];  // (2048, 4096)
    const float* Ro    = (const float*)d_in[3];  // (2048, 4096)
    const float* theta = (const float*)d_in[4];  // (69,)
    float* out = (float*)d_out;                  // (2048,)

    float* ws   = (float*)d_ws;
    float* bo   = ws;                 // 4096*4
    float* bi   = ws + EE * 4;        // 4096*4
    float* Mbuf = ws + 2 * EE * 4;    // 2048*8 (mi | mo)

    gemm_bt_kernel<<<dim3(EE / 16, 2), 32, 0, stream>>>(Ro, Ri, X, bo, bi);
    gemm_m_kernel<<<dim3(NN / 16, 2), 32, 0, stream>>>(Ri, Ro, evec, bo, bi, Mbuf);
    sim_kernel<<<NN, 256, 0, stream>>>(Mbuf, X, theta, out);
}